// priMaskST_5360119186109
// MI455X (gfx1250) — compile-verified
//
#include <hip/hip_runtime.h>
#include <hip/hip_bf16.h>
#include <math.h>

typedef __attribute__((ext_vector_type(2))) float v2f;
typedef __attribute__((ext_vector_type(8))) float v8f;

#define Bn 64
#define Ln 12
#define Tn 5000
#define Cn 23
#define TTF 312                 // full 16-wide tiles per batch
#define NFULL (Bn * TTF)        // 19968 full tiles
#define TAIL_T0 (TTF * 16)      // 4992
#define NB1 64                  // stats-pass blocks

// Workspace layout (floats):
// [0..255]    A16 padded 16x16 (row-major, zeros outside 12x12)
// [256..271]  sc2 per channel (x16, padded 0)
// [272..287]  bs2 per channel (x16, padded 0)
// [288]       p0,  [289] p1
// [320..320+NB1*32)  block partials (per block: 16 sums, 16 sumsqs)

// One 16(leads) x 16(timesteps) fp32 tile via 3 chained V_WMMA_F32_16X16X4_F32.
// A-frag: lane l holds A16[l][k0+2*hi .. +1]; B-frag: lane l holds x[i][t0+l].
template <bool GUARD>
__device__ __forceinline__ v8f wmma_tile(const float* __restrict__ xb, int t,
                                         bool tvalid, const v2f a[3], int hi) {
  v8f c = {0, 0, 0, 0, 0, 0, 0, 0};
#pragma unroll
  for (int k = 0; k < 3; ++k) {
    int i0 = 4 * k + 2 * hi;
    v2f bf;
    if (GUARD) {
      bf.x = tvalid ? xb[i0 * Tn + t] : 0.0f;
      bf.y = tvalid ? xb[(i0 + 1) * Tn + t] : 0.0f;
    } else {
      bf.x = xb[i0 * Tn + t];
      bf.y = xb[(i0 + 1) * Tn + t];
    }
    c = __builtin_amdgcn_wmma_f32_16x16x4_f32(false, a[k], false, bf,
                                              (short)0, c, false, false);
  }
  return c;
}

// ---------------- Stage 0: tiny prolog (A matrix, gates) ----------------
__global__ void priMaskST_stage0(const float* __restrict__ matrix,
                                 const float* __restrict__ mask,
                                 const float* __restrict__ conv_w,
                                 const float* __restrict__ merge_w,
                                 const float* __restrict__ merge_b,
                                 float* __restrict__ ws) {
  __shared__ float colsum[Cn];
  int tid = threadIdx.x;
  ws[tid] = 0.0f;  // zero A16 region (blockDim == 256)
  if (tid < Cn) {
    float s = 0.0f;
    for (int i = 0; i < Ln; ++i) s += matrix[i * Cn + tid];
    colsum[tid] = s;
  }
  if (tid == 0) {
    ws[288] = 1.0f / (1.0f + expf(-(merge_w[0] + merge_b[0])));
    ws[289] = 1.0f / (1.0f + expf(-(merge_w[1] + merge_b[1])));
  }
  __syncthreads();
  if (tid < Ln * Ln) {
    int o = tid / Ln, i = tid - o * Ln;
    float a = 0.0f;
    for (int c = 0; c < Cn; ++c) {
      float me = fmaxf(matrix[i * Cn + c] / colsum[c], 0.0f) * mask[i * Cn + c];
      a += conv_w[(o * Ln + i) * Cn + c] * me;
    }
    ws[o * 16 + i] = a;
  }
}

// ---------------- Pass 1: WMMA stats (sum d, sum d^2 per channel) --------
__global__ void priMaskST_pass1(const float* __restrict__ x,
                                const float* __restrict__ ws,
                                float* __restrict__ part) {
  const int lane = threadIdx.x & 31;
  const int l = lane & 15;
  const int hi = (lane >> 4) & 1;
  const int waveInBlock = threadIdx.x >> 5;
  const int gwave = blockIdx.x * (blockDim.x >> 5) + waveInBlock;
  const int nWaves = gridDim.x * (blockDim.x >> 5);

  v2f a[3];
#pragma unroll
  for (int k = 0; k < 3; ++k) {
    int k0 = 4 * k + 2 * hi;
    a[k].x = ws[l * 16 + k0];
    a[k].y = ws[l * 16 + k0 + 1];
  }

  float s[8] = {0, 0, 0, 0, 0, 0, 0, 0};
  float q[8] = {0, 0, 0, 0, 0, 0, 0, 0};

  // full tiles: unguarded loads, full-EXEC everywhere
  for (int tile = gwave; tile < NFULL; tile += nWaves) {
    int b = tile / TTF;
    int t0 = (tile - b * TTF) * 16;
    const float* xb = x + (size_t)b * (Ln * Tn);
    v8f c = wmma_tile<false>(xb, t0 + l, true, a, hi);
#pragma unroll
    for (int r = 0; r < 8; ++r) {
      s[r] += c[r];
      q[r] += c[r] * c[r];
    }
  }
  // tail tile (t0=4992, 8 valid timesteps): one per wave for the first 64 waves
  if (gwave < Bn) {
    const float* xb = x + (size_t)gwave * (Ln * Tn);
    bool tvalid = (TAIL_T0 + l) < Tn;
    v8f c = wmma_tile<true>(xb, TAIL_T0 + l, tvalid, a, hi);
#pragma unroll
    for (int r = 0; r < 8; ++r) {
      float dv = tvalid ? c[r] : 0.0f;
      s[r] += dv;
      q[r] += dv * dv;
    }
  }

  // deterministic xor-tree reduction within each 16-lane half
#pragma unroll
  for (int r = 0; r < 8; ++r) {
    float sv = s[r], qv = q[r];
#pragma unroll
    for (int m = 1; m < 16; m <<= 1) {
      sv += __shfl_xor(sv, m, 32);
      qv += __shfl_xor(qv, m, 32);
    }
    s[r] = sv;
    q[r] = qv;
  }

  __shared__ float ldsS[8][16];
  __shared__ float ldsQ[8][16];
  if (l == 0) {  // lanes 0 (hi=0 -> ch r) and 16 (hi=1 -> ch r+8)
#pragma unroll
    for (int r = 0; r < 8; ++r) {
      ldsS[waveInBlock][r + 8 * hi] = s[r];
      ldsQ[waveInBlock][r + 8 * hi] = q[r];
    }
  }
  __syncthreads();
  if (threadIdx.x < 16) {
    float S = 0.0f, Q = 0.0f;
    for (int w = 0; w < 8; ++w) {
      S += ldsS[w][threadIdx.x];
      Q += ldsQ[w][threadIdx.x];
    }
    part[blockIdx.x * 32 + threadIdx.x] = S;
    part[blockIdx.x * 32 + 16 + threadIdx.x] = Q;
  }
}

// ---------------- Finalize: fold BN + gates into per-channel scale/bias --
__global__ void priMaskST_finalize(const float* __restrict__ gamma,
                                   const float* __restrict__ beta,
                                   const float* __restrict__ part,
                                   float* __restrict__ ws) {
  int j = threadIdx.x;
  if (j < 16) {
    float S = 0.0f, Q = 0.0f;
    for (int b = 0; b < NB1; ++b) {
      S += part[b * 32 + j];
      Q += part[b * 32 + 16 + j];
    }
    float sc2 = 0.0f, bs2 = 0.0f;
    if (j < Ln) {
      const float N = (float)Bn * (float)Tn;
      float mean_d = S / N;
      float var = fmaxf(Q / N - mean_d * mean_d, 0.0f);
      float rstd = rsqrtf(var + 1e-5f);
      float sc = rstd * gamma[j];
      float p0 = ws[288];
      sc2 = sc * p0;
      bs2 = (beta[j] - mean_d * sc) * p0;
    }
    ws[256 + j] = sc2;
    ws[272 + j] = bs2;
  }
}

// ---------------- Pass 2: WMMA compute + normalize + gate + residual -----
__global__ void priMaskST_pass2(const float* __restrict__ x,
                                const float* __restrict__ ws,
                                float* __restrict__ out) {
  const int lane = threadIdx.x & 31;
  const int l = lane & 15;
  const int hi = (lane >> 4) & 1;
  const int gwave = blockIdx.x * (blockDim.x >> 5) + (threadIdx.x >> 5);
  const int nWaves = gridDim.x * (blockDim.x >> 5);

  v2f a[3];
#pragma unroll
  for (int k = 0; k < 3; ++k) {
    int k0 = 4 * k + 2 * hi;
    a[k].x = ws[l * 16 + k0];
    a[k].y = ws[l * 16 + k0 + 1];
  }
  float sc2[8], bs2[8];
#pragma unroll
  for (int r = 0; r < 8; ++r) {
    int o = r + 8 * hi;
    sc2[r] = ws[256 + o];
    bs2[r] = ws[272 + o];
  }
  const float p1 = ws[289];

  // full tiles: unguarded loads; store mask only on padded rows (r>=4, hi=1)
  for (int tile = gwave; tile < NFULL; tile += nWaves) {
    int b = tile / TTF;
    int t0 = (tile - b * TTF) * 16;
    const float* xb = x + (size_t)b * (Ln * Tn);
    int t = t0 + l;
    v8f c = wmma_tile<false>(xb, t, true, a, hi);
    float* ob = out + (size_t)b * (Ln * Tn);
#pragma unroll
    for (int r = 0; r < 8; ++r) {
      if (r < 4 || hi == 0) {  // o = r + 8*hi < 12
        int o = r + 8 * hi;
        float yn = c[r] * sc2[r] + bs2[r];
        ob[o * Tn + t] = fmaxf(yn, 0.0f) + xb[o * Tn + t] * p1;
      }
    }
  }
  // tail tile per batch
  if (gwave < Bn) {
    const float* xb = x + (size_t)gwave * (Ln * Tn);
    int t = TAIL_T0 + l;
    bool tvalid = t < Tn;
    v8f c = wmma_tile<true>(xb, t, tvalid, a, hi);
    float* ob = out + (size_t)gwave * (Ln * Tn);
#pragma unroll
    for (int r = 0; r < 8; ++r) {
      int o = r + 8 * hi;
      if ((r < 4 || hi == 0) && tvalid) {
        float yn = c[r] * sc2[r] + bs2[r];
        ob[o * Tn + t] = fmaxf(yn, 0.0f) + xb[o * Tn + t] * p1;
      }
    }
  }
}

extern "C" void kernel_launch(void* const* d_in, const int* in_sizes, int n_in,
                              void* d_out, int out_size, void* d_ws, size_t ws_size,
                              hipStream_t stream) {
  const float* x       = (const float*)d_in[0];
  const float* matrix  = (const float*)d_in[1];
  const float* mask    = (const float*)d_in[2];
  const float* conv_w  = (const float*)d_in[3];
  // d_in[4] = conv_b (cancels out of the BN-normalized expression)
  const float* gamma   = (const float*)d_in[5];
  const float* beta    = (const float*)d_in[6];
  const float* merge_w = (const float*)d_in[7];
  const float* merge_b = (const float*)d_in[8];
  float* out = (float*)d_out;
  float* ws  = (float*)d_ws;
  float* part = ws + 320;

  priMaskST_stage0<<<dim3(1), dim3(256), 0, stream>>>(matrix, mask, conv_w,
                                                      merge_w, merge_b, ws);
  priMaskST_pass1<<<dim3(NB1), dim3(256), 0, stream>>>(x, ws, part);
  priMaskST_finalize<<<dim3(1), dim3(32), 0, stream>>>(gamma, beta, part, ws);
  priMaskST_pass2<<<dim3(512), dim3(256), 0, stream>>>(x, ws, out);
}